// SimpleMPS_53025666237058
// MI455X (gfx1250) — compile-verified
//
#include <hip/hip_runtime.h>
#include <math.h>

typedef __attribute__((ext_vector_type(2))) float v2f;
typedef __attribute__((ext_vector_type(8))) float v8f;

#define N_SITES 64
#define BOND    32
#define PADK    34                    // padded K-stride (even -> 8B-aligned pairs, bank-spread)
#define SELSTR  (BOND * PADK)         // 1088 floats per phys matrix
#define WAVES_PER_BLOCK 8
#define ROWS_PER_WAVE   16
#define ROWS_PER_BLOCK  (WAVES_PER_BLOCK * ROWS_PER_WAVE)   // 128

// Batched MPS amplitude chain on V_WMMA_F32_16X16X4_F32.
// Per wave: 16(batch) x 32(bond) state tile, image kept in LDS.
// Per site: S_new = (S*mask0) @ M0 + (S*mask1) @ M1   (per-row spin select, no divergence)
// Matrices staged in LDS *transposed* (MT[sel][n][k]) so each WMMA B operand is one ds_load_b64.
// Double-buffered staging, software-pipelined: one barrier per site, global loads overlap WMMAs.
__global__ __launch_bounds__(256) void SimpleMPS_wmma_kernel(
    const float* __restrict__ t_first,   // (2,1,32)
    const float* __restrict__ t_mid,     // (62,2,32,32)
    const float* __restrict__ t_last,    // (2,32,1)
    const int*   __restrict__ x,         // (32768,64)  values in {-1,+1}
    float*       __restrict__ out)       // (32768,)
{
    __shared__ __align__(16) float ldsMT[2][2 * SELSTR];                     // ~17.4 KB (double-buffered, transposed)
    __shared__ __align__(16) float ldsS[WAVES_PER_BLOCK][ROWS_PER_WAVE * BOND];  // 16 KB state images

    const int tid    = threadIdx.x;
    const int wid    = tid >> 5;
    const int lane   = tid & 31;
    const int laneLo = lane & 15;     // batch row (A) / output column n (B,C)
    const int laneHi = lane >> 4;     // selects the K pair / M half
    const int rowBase = blockIdx.x * ROWS_PER_BLOCK + wid * ROWS_PER_WAVE;

    float* S = &ldsS[wid][0];

    // staging-thread coordinates: thread t owns global dwords [4t..4t+3] of each 32x32 matrix
    const int stK  = tid >> 3;         // matrix row k (contiguous-n float4)
    const int stN0 = (tid & 7) * 4;    // starting column n

    // ---- init state: S[r][c] = t_first[idx(x[r,0])][0][c]
    {
        const int r  = laneLo;
        const int cb = laneHi * 16;
        const int p  = (x[(rowBase + r) * N_SITES + 0] + 1) >> 1;
        const float4* src = (const float4*)(t_first + p * BOND + cb);
        float4* dst = (float4*)(S + r * BOND + cb);
        dst[0] = src[0]; dst[1] = src[1]; dst[2] = src[2]; dst[3] = src[3];
    }

    // ---- prologue: stage site 1 matrices (mi=0) into buffer 0, transposed
    {
        const float4* g = (const float4*)(t_mid);
        float4 ra = g[tid];         // sel 0
        float4 rb = g[tid + 256];   // sel 1
        float* d0 = &ldsMT[0][stK];
        float* d1 = &ldsMT[0][SELSTR + stK];
        d0[(stN0 + 0) * PADK] = ra.x; d0[(stN0 + 1) * PADK] = ra.y;
        d0[(stN0 + 2) * PADK] = ra.z; d0[(stN0 + 3) * PADK] = ra.w;
        d1[(stN0 + 0) * PADK] = rb.x; d1[(stN0 + 1) * PADK] = rb.y;
        d1[(stN0 + 2) * PADK] = rb.z; d1[(stN0 + 3) * PADK] = rb.w;
    }

    for (int site = 1; site <= N_SITES - 2; ++site) {
        const int mi  = site - 1;
        const int cur = (site - 1) & 1;
        const int nxt = site & 1;

        // pipeline: issue next site's global loads before the barrier / compute
        float4 ra, rb;
        const bool havenext = (site < N_SITES - 2);
        if (havenext) {
            const float4* g = (const float4*)(t_mid + (size_t)(mi + 1) * 2 * BOND * BOND);
            ra = g[tid];
            rb = g[tid + 256];
        }
        if (site + 1 < N_SITES - 2) {   // prefetch site+2's 8KB slab across the block
            __builtin_prefetch(t_mid + (size_t)(mi + 2) * 2 * BOND * BOND + tid * 8, 0, 1);
        }

        __syncthreads();   // buf[cur] fully staged (prev iter's stores / prologue)

        // per-lane row-selection masks for this site (row m = laneLo)
        const int xv = x[(rowBase + laneLo) * N_SITES + site];
        const float m1 = (xv > 0) ? 1.0f : 0.0f;   // occupancy idx == 1
        const float m0 = 1.0f - m1;                // occupancy idx == 0

        const float* MT = &ldsMT[cur][0];

        v8f acc0 = {0.f,0.f,0.f,0.f,0.f,0.f,0.f,0.f};  // N = 0..15
        v8f acc1 = {0.f,0.f,0.f,0.f,0.f,0.f,0.f,0.f};  // N = 16..31

        #pragma unroll
        for (int kb = 0; kb < 8; ++kb) {
            const int k0 = kb * 4 + laneHi * 2;   // this lane's K pair in the 16x4 chunk

            // A: contiguous state pair (ds_load_b64), masked per selection
            const v2f av = *(const v2f*)(S + laneLo * BOND + k0);
            v2f a0; a0.x = av.x * m0; a0.y = av.y * m0;
            v2f a1; a1.x = av.x * m1; a1.y = av.y * m1;

            // B: transposed LDS -> each operand is one aligned contiguous b64
            const v2f b00 = *(const v2f*)(MT + (laneLo     ) * PADK + k0);           // sel0, N=0..15
            const v2f b01 = *(const v2f*)(MT + (laneLo + 16) * PADK + k0);           // sel0, N=16..31
            const v2f b10 = *(const v2f*)(MT + SELSTR + (laneLo     ) * PADK + k0);  // sel1, N=0..15
            const v2f b11 = *(const v2f*)(MT + SELSTR + (laneLo + 16) * PADK + k0);  // sel1, N=16..31

            acc0 = __builtin_amdgcn_wmma_f32_16x16x4_f32(false, a0, false, b00, (short)0, acc0, false, false);
            acc0 = __builtin_amdgcn_wmma_f32_16x16x4_f32(false, a1, false, b10, (short)0, acc0, false, false);
            acc1 = __builtin_amdgcn_wmma_f32_16x16x4_f32(false, a0, false, b01, (short)0, acc1, false, false);
            acc1 = __builtin_amdgcn_wmma_f32_16x16x4_f32(false, a1, false, b11, (short)0, acc1, false, false);
        }

        // write S_new back to the state image: D[m][n], m = v + 8*laneHi, n = laneLo (+16)
        #pragma unroll
        for (int v = 0; v < 8; ++v) {
            const int m = v + 8 * laneHi;
            S[m * BOND + laneLo]      = acc0[v];
            S[m * BOND + 16 + laneLo] = acc1[v];
        }

        // stage next site's matrices into the other buffer (read at top of next iter, after barrier)
        if (havenext) {
            float* d0 = &ldsMT[nxt][stK];
            float* d1 = &ldsMT[nxt][SELSTR + stK];
            d0[(stN0 + 0) * PADK] = ra.x; d0[(stN0 + 1) * PADK] = ra.y;
            d0[(stN0 + 2) * PADK] = ra.z; d0[(stN0 + 3) * PADK] = ra.w;
            d1[(stN0 + 0) * PADK] = rb.x; d1[(stN0 + 1) * PADK] = rb.y;
            d1[(stN0 + 2) * PADK] = rb.z; d1[(stN0 + 3) * PADK] = rb.w;
        }
    }

    // ---- finalize: amp[r] = sum_c S[r][c] * t_last[idx(x[r,63])][c]; out = log(amp)
    {
        const int r  = laneLo;
        const int cb = laneHi * 16;
        const int p  = (x[(rowBase + r) * N_SITES + (N_SITES - 1)] + 1) >> 1;
        float partial = 0.0f;
        #pragma unroll
        for (int j = 0; j < 16; ++j)
            partial += S[r * BOND + cb + j] * t_last[p * BOND + cb + j];
        partial += __shfl_xor(partial, 16, 32);
        if (laneHi == 0)
            out[rowBase + r] = logf(partial);
    }
}

extern "C" void kernel_launch(void* const* d_in, const int* in_sizes, int n_in,
                              void* d_out, int out_size, void* d_ws, size_t ws_size,
                              hipStream_t stream) {
    const float* t_first = (const float*)d_in[0];   // (2,1,32)
    const float* t_mid   = (const float*)d_in[1];   // (62,2,32,32)
    const float* t_last  = (const float*)d_in[2];   // (2,32,1)
    const int*   x       = (const int*)  d_in[3];   // (32768,64)
    float* out = (float*)d_out;                     // (32768,)

    const int batch  = in_sizes[3] / N_SITES;       // 32768
    const int blocks = batch / ROWS_PER_BLOCK;      // 256
    SimpleMPS_wmma_kernel<<<blocks, 256, 0, stream>>>(t_first, t_mid, t_last, x, out);
}